// NaiveAttention_79242146611392
// MI455X (gfx1250) — compile-verified
//
#include <hip/hip_runtime.h>
#include <cmath>

typedef __attribute__((ext_vector_type(2))) float v2f;
typedef __attribute__((ext_vector_type(8))) float v8f;

static constexpr int H_  = 16;
static constexpr int HD_ = 128;
static constexpr int D_  = 2048;
static constexpr int S_  = 2048;
static constexpr int B_  = 2;

__device__ __forceinline__ v8f wmma4(v2f a, v2f b, v8f c) {
  // D = A(16x4,f32) * B(4x16,f32) + C(16x16,f32)
  return __builtin_amdgcn_wmma_f32_16x16x4_f32(false, a, false, b, (short)0, c,
                                               false, false);
}

// C[m,n] = alpha * sum_k A[m,k]*B[n,k]  (+ bias[n]) (+ addm[m,n])
// One wave computes a 64x64 tile: 4x4 grid of 16x16 WMMA accumulators.
// NT_C: use non-temporal stores for C (streaming output that exceeds L2).
template <bool NT_C>
__global__ __launch_bounds__(32)
void gemm_abt(const float* __restrict__ A, int lda, long sAb, long sAh,
              const float* __restrict__ Bt, int ldb, long sBb, long sBh,
              float* __restrict__ C, int ldc, long sCb, long sCh,
              const float* __restrict__ bias,
              const float* __restrict__ addm, int ldadd,
              int K, float alpha, int nh) {
  const int lane = threadIdx.x;
  const int l15  = lane & 15;
  const int kh   = (lane >> 4) * 2;        // K sub-offset: 0 or 2
  const int n0   = blockIdx.x * 64;
  const int m0   = blockIdx.y * 64;
  const int z    = blockIdx.z;
  const int bz   = z / nh, hz = z % nh;
  A  += (long)bz * sAb + (long)hz * sAh;
  Bt += (long)bz * sBb + (long)hz * sBh;
  C  += (long)bz * sCb + (long)hz * sCh;

  v8f acc[4][4] = {};

  for (int k = 0; k < K; k += 4) {
    v2f af[4], bf[4];
#pragma unroll
    for (int i = 0; i < 4; ++i)
      af[i] = *(const v2f*)(A + (long)(m0 + 16 * i + l15) * lda + (k + kh));
#pragma unroll
    for (int j = 0; j < 4; ++j)
      bf[j] = *(const v2f*)(Bt + (long)(n0 + 16 * j + l15) * ldb + (k + kh));
#pragma unroll
    for (int i = 0; i < 4; ++i)
#pragma unroll
      for (int j = 0; j < 4; ++j)
        acc[i][j] = wmma4(af[i], bf[j], acc[i][j]);
  }

  const int mh = (lane >> 4) * 8;
#pragma unroll
  for (int i = 0; i < 4; ++i) {
#pragma unroll
    for (int j = 0; j < 4; ++j) {
      const int n = n0 + 16 * j + l15;
      const float bb = bias ? bias[n] : 0.0f;
#pragma unroll
      for (int r = 0; r < 8; ++r) {
        const int m = m0 + 16 * i + mh + r;
        float v = acc[i][j][r] * alpha + bb;
        if (addm) v += addm[(long)m * ldadd + n];  // mask: hot in L2, RT load
        if (NT_C)
          __builtin_nontemporal_store(v, &C[(long)m * ldc + n]);
        else
          C[(long)m * ldc + n] = v;
      }
    }
  }
}

// C[m,n] = sum_k A[m,k]*B[k,n]   (B row-major K x N); A is streaming (attn).
__global__ __launch_bounds__(32)
void gemm_ab(const float* __restrict__ A, int lda, long sAb, long sAh,
             const float* __restrict__ Bm, int ldb, long sBb, long sBh,
             float* __restrict__ C, int ldc, long sCb, long sCh,
             int K, int nh) {
  const int lane = threadIdx.x;
  const int l15  = lane & 15;
  const int kh   = (lane >> 4) * 2;
  const int n0   = blockIdx.x * 64;
  const int m0   = blockIdx.y * 64;
  const int z    = blockIdx.z;
  const int bz   = z / nh, hz = z % nh;
  A  += (long)bz * sAb + (long)hz * sAh;
  Bm += (long)bz * sBb + (long)hz * sBh;
  C  += (long)bz * sCb + (long)hz * sCh;

  v8f acc[4][4] = {};

  for (int k = 0; k < K; k += 4) {
    v2f af[4], bf[4];
#pragma unroll
    for (int i = 0; i < 4; ++i)
      af[i] = __builtin_nontemporal_load(
          (const v2f*)(A + (long)(m0 + 16 * i + l15) * lda + (k + kh)));
#pragma unroll
    for (int j = 0; j < 4; ++j) {
      const float* bp = Bm + (long)(k + kh) * ldb + (n0 + 16 * j + l15);
      bf[j] = (v2f){bp[0], bp[ldb]};           // V: hot in L2, RT loads
    }
#pragma unroll
    for (int i = 0; i < 4; ++i)
#pragma unroll
      for (int j = 0; j < 4; ++j)
        acc[i][j] = wmma4(af[i], bf[j], acc[i][j]);
  }

  const int mh = (lane >> 4) * 8;
#pragma unroll
  for (int i = 0; i < 4; ++i)
#pragma unroll
    for (int j = 0; j < 4; ++j) {
      const int n = n0 + 16 * j + l15;
#pragma unroll
      for (int r = 0; r < 8; ++r) {
        const int m = m0 + 16 * i + mh + r;
        C[(long)m * ldc + n] = acc[i][j][r];   // ctx: re-read by out-proj, RT
      }
    }
}

// In-place row softmax over S_ columns; one 256-thread block per row.
// attn is streaming (3.4x L2 size): non-temporal in both directions.
__global__ __launch_bounds__(256)
void softmax_rows(float* __restrict__ attn) {
  const long row = blockIdx.x;
  float* p = attn + row * (long)S_;
  const int t = threadIdx.x;
  float v[8];
  float mx = -3.402823466e38f;
#pragma unroll
  for (int i = 0; i < 8; ++i) {
    v[i] = __builtin_nontemporal_load(&p[t + 256 * i]);
    mx = fmaxf(mx, v[i]);
  }
#pragma unroll
  for (int off = 16; off > 0; off >>= 1) mx = fmaxf(mx, __shfl_xor(mx, off, 32));
  __shared__ float smax[8];
  __shared__ float ssum[8];
  if ((t & 31) == 0) smax[t >> 5] = mx;
  __syncthreads();
#pragma unroll
  for (int i = 0; i < 8; ++i) mx = fmaxf(mx, smax[i]);
  float sum = 0.0f;
#pragma unroll
  for (int i = 0; i < 8; ++i) { v[i] = expf(v[i] - mx); sum += v[i]; }
#pragma unroll
  for (int off = 16; off > 0; off >>= 1) sum += __shfl_xor(sum, off, 32);
  if ((t & 31) == 0) ssum[t >> 5] = sum;
  __syncthreads();
  sum = 0.0f;
#pragma unroll
  for (int i = 0; i < 8; ++i) sum += ssum[i];
  const float inv = 1.0f / sum;
#pragma unroll
  for (int i = 0; i < 8; ++i)
    __builtin_nontemporal_store(v[i] * inv, &p[t + 256 * i]);
}

extern "C" void kernel_launch(void* const* d_in, const int* in_sizes, int n_in,
                              void* d_out, int out_size, void* d_ws,
                              size_t ws_size, hipStream_t stream) {
  const float* X    = (const float*)d_in[0];
  const float* mask = (const float*)d_in[1];
  const float* Wq   = (const float*)d_in[2];
  const float* bq   = (const float*)d_in[3];
  const float* Wk   = (const float*)d_in[4];
  const float* bk   = (const float*)d_in[5];
  const float* Wv   = (const float*)d_in[6];
  const float* bv   = (const float*)d_in[7];
  const float* Wo   = (const float*)d_in[8];
  const float* bo   = (const float*)d_in[9];

  float* out  = (float*)d_out;
  const long BSD = (long)B_ * S_ * D_;
  float* attn = out + BSD;  // outputs concatenated: out then attn

  float* Q   = (float*)d_ws;
  float* Km  = Q + BSD;
  float* V   = Km + BSD;
  float* CTX = V + BSD;

  dim3 w(32);

  // --- QKV projections: [4096 x 2048] = X @ W^T + b ---
  dim3 gp(D_ / 64, (B_ * S_) / 64, 1);
  gemm_abt<false><<<gp, w, 0, stream>>>(X, D_, 0, 0, Wq, D_, 0, 0, Q, D_, 0, 0,
                                        bq, nullptr, 0, D_, 1.0f, 1);
  gemm_abt<false><<<gp, w, 0, stream>>>(X, D_, 0, 0, Wk, D_, 0, 0, Km, D_, 0, 0,
                                        bk, nullptr, 0, D_, 1.0f, 1);
  gemm_abt<false><<<gp, w, 0, stream>>>(X, D_, 0, 0, Wv, D_, 0, 0, V, D_, 0, 0,
                                        bv, nullptr, 0, D_, 1.0f, 1);

  // --- scores = Q_h @ K_h^T / sqrt(hd) + mask  (NT store into attn region) ---
  dim3 gs(S_ / 64, S_ / 64, B_ * H_);
  const float sc = 1.0f / sqrtf((float)HD_);
  gemm_abt<true><<<gs, w, 0, stream>>>(Q, D_, (long)S_ * D_, HD_,
                                       Km, D_, (long)S_ * D_, HD_,
                                       attn, S_, (long)H_ * S_ * S_,
                                       (long)S_ * S_,
                                       nullptr, mask, S_, HD_, sc, H_);

  // --- softmax rows in place (NT both directions) ---
  softmax_rows<<<dim3(B_ * H_ * S_), dim3(256), 0, stream>>>(attn);

  // --- ctx = attn @ V_h  (NT loads of attn) ---
  dim3 gv(HD_ / 64, S_ / 64, B_ * H_);
  gemm_ab<<<gv, w, 0, stream>>>(attn, S_, (long)H_ * S_ * S_, (long)S_ * S_,
                                V, D_, (long)S_ * D_, HD_,
                                CTX, D_, (long)S_ * D_, HD_, S_, H_);

  // --- out = ctx @ Wo^T + bo ---
  gemm_abt<false><<<gp, w, 0, stream>>>(CTX, D_, 0, 0, Wo, D_, 0, 0, out, D_,
                                        0, 0, bo, nullptr, 0, D_, 1.0f, 1);
}